// semantic_chamfer_distance_68702296866985
// MI455X (gfx1250) — compile-verified
//
#include <hip/hip_runtime.h>

typedef float v2f __attribute__((ext_vector_type(2)));
typedef float v8f __attribute__((ext_vector_type(8)));

// Shapes fixed by the reference: B=2, S=4, N=M=4096, 3-D points.
constexpr int Bdim = 2;
constexpr int Sdim = 4;
constexpr int Npts = 4096;
constexpr int Mpts = 4096;

constexpr int WAVES          = 8;           // 8 wave32 per block
constexpr int BLOCK          = WAVES * 32;  // 256 threads
constexpr int ROWS_PER_BLOCK = WAVES * 16;  // each wave owns a 16-row tile

// Zero the 16-float accumulator workspace (8 per direction).
__global__ void chamfer_init_kernel(float* ws) {
    if (threadIdx.x < 16) ws[threadIdx.x] = 0.0f;
}

// One direction of Chamfer: for each row point of A, min squared distance over
// all points of Bset; per-(b,s) sum of those mins is atomically accumulated.
//   A: [BS, nA, 3]   Bset: [BS, nB, 3]   accum: [BS]
__global__ __launch_bounds__(BLOCK)
void chamfer_dir_kernel(const float* __restrict__ Aall,
                        const float* __restrict__ Ball,
                        float* __restrict__ accum,
                        int nA, int nB)
{
    // Stage target points + their squared norms in LDS: 4096 * float4 = 64 KB.
    __shared__ float4 ycol[Mpts];

    const int bs = blockIdx.y;
    const float* __restrict__ Ap = Aall + (size_t)bs * nA * 3;
    const float* __restrict__ Bp = Ball + (size_t)bs * nB * 3;

    const int tid = threadIdx.x;
    for (int i = tid; i < nB; i += BLOCK) {
        float px = Bp[3 * i + 0];
        float py = Bp[3 * i + 1];
        float pz = Bp[3 * i + 2];
        ycol[i] = make_float4(px, py, pz, px * px + py * py + pz * pz);
    }
    __syncthreads();

    const int lane = tid & 31;
    const int wave = tid >> 5;
    const int half = lane >> 4;   // 0: K={0,1}, 1: K={2,3}
    const int l16  = lane & 15;

    // A-matrix fragment: row (lane&15) of this wave's 16-row tile.
    const int row = blockIdx.x * ROWS_PER_BLOCK + wave * 16 + l16;
    const float xr0 = Ap[3 * row + 0];
    const float xr1 = Ap[3 * row + 1];
    const float xr2 = Ap[3 * row + 2];
    const float x2r = xr0 * xr0 + xr1 * xr1 + xr2 * xr2;

    v2f a;
    a.x = half ? xr2  : xr0;   // K = 0 or 2
    a.y = half ? 0.0f : xr1;   // K = 1 or 3 (pad)

    // Running per-row mins of (y^2 - 2*xy); this lane's column slice.
    float rmin[8];
#pragma unroll
    for (int j = 0; j < 8; ++j) rmin[j] = 3.402823466e38f;

    const int ntiles = nB / 16;
    for (int t = 0; t < ntiles; ++t) {
        const float4 yc = ycol[t * 16 + l16];   // column this lane owns in C/D
        v2f bb;
        bb.x = half ? yc.z : yc.x;
        bb.y = half ? 0.0f : yc.y;

        v8f acc = {};
        // D = A(16x4) * B(4x16) + 0  ->  xy tile
        acc = __builtin_amdgcn_wmma_f32_16x16x4_f32(
            /*neg_a=*/false, a, /*neg_b=*/false, bb,
            /*c_mod=*/(short)0, acc, /*reuse_a=*/false, /*reuse_b=*/false);

#pragma unroll
        for (int j = 0; j < 8; ++j)
            rmin[j] = fminf(rmin[j], yc.w - 2.0f * acc[j]);
    }

    // Min-reduce across the 16 lanes of each half (columns 0..15).
#pragma unroll
    for (int m = 1; m < 16; m <<= 1) {
#pragma unroll
        for (int j = 0; j < 8; ++j)
            rmin[j] = fminf(rmin[j], __shfl_xor(rmin[j], m, 32));
    }

    // Sum the 8 row-mins held by this half (rows 0-7 in half 0, 8-15 in half 1).
    float rowsum = 0.0f;
#pragma unroll
    for (int j = 0; j < 8; ++j) rowsum += rmin[j];

    // Sum of x^2 over all 16 rows of the tile (identical in both halves).
    float x2sum = x2r;
#pragma unroll
    for (int m = 1; m < 16; m <<= 1) x2sum += __shfl_xor(x2sum, m, 32);

    // Combine halves: total = sum over 16 rows of (x2[m] + min_n(y2 - 2xy)).
    const float total = rowsum + __shfl_xor(rowsum, 16, 32) + x2sum;

    if (lane == 0) atomicAdd(&accum[bs], total);
}

// sums[0..7]  = per-(b,s) sum of dist1 over N   (bs = b*S + s)
// sums[8..15] = per-(b,s) sum of dist2 over M
// out[0..S-1]        = chamfer_distances[s]      (mean over b)
// out[S .. S+S*B-1]  = chamfer_distances_tensor  [S,B] row-major
__global__ void chamfer_finalize_kernel(const float* __restrict__ sums,
                                        float* __restrict__ out)
{
    const int tid = threadIdx.x;
    if (tid < Sdim * Bdim) {
        const int s  = tid / Bdim;
        const int b  = tid % Bdim;
        const int bs = b * Sdim + s;
        const float comb = sums[bs] / (float)Npts + sums[8 + bs] / (float)Mpts;
        out[Sdim + s * Bdim + b] = comb;
    }
    __syncthreads();
    if (tid < Sdim) {
        float acc = 0.0f;
        for (int b = 0; b < Bdim; ++b) {
            const int bs = b * Sdim + tid;
            acc += sums[bs] / (float)Npts + sums[8 + bs] / (float)Mpts;
        }
        out[tid] = acc / (float)Bdim;
    }
}

extern "C" void kernel_launch(void* const* d_in, const int* in_sizes, int n_in,
                              void* d_out, int out_size, void* d_ws, size_t ws_size,
                              hipStream_t stream) {
    const float* outp = (const float*)d_in[0];   // output_points [B,S,N,3]
    const float* tgtp = (const float*)d_in[1];   // target_points [B,S,M,3]
    float* ws  = (float*)d_ws;                   // 16 accumulators
    float* out = (float*)d_out;                  // 4 + 8 floats

    chamfer_init_kernel<<<1, 32, 0, stream>>>(ws);

    dim3 block(BLOCK);
    dim3 grid1(Npts / ROWS_PER_BLOCK, Bdim * Sdim);   // dist1: out -> tgt
    chamfer_dir_kernel<<<grid1, block, 0, stream>>>(outp, tgtp, ws, Npts, Mpts);

    dim3 grid2(Mpts / ROWS_PER_BLOCK, Bdim * Sdim);   // dist2: tgt -> out
    chamfer_dir_kernel<<<grid2, block, 0, stream>>>(tgtp, outp, ws + 8, Mpts, Npts);

    chamfer_finalize_kernel<<<1, 32, 0, stream>>>(ws, out);
}